// MultiHeadAttention_26731876450935
// MI455X (gfx1250) — compile-verified
//
#include <hip/hip_runtime.h>
#include <math.h>

// Problem constants (fixed by the reference)
constexpr int Bc  = 4;
constexpr int Sc  = 1024;
constexpr int Dc  = 1024;
constexpr int Hc  = 16;
constexpr int DKc = 64;   // D / H

typedef float v2f __attribute__((ext_vector_type(2)));
typedef float v8f __attribute__((ext_vector_type(8)));
typedef int   v4i __attribute__((ext_vector_type(4)));

// CDNA5 async global->LDS path (guarded: falls back to sync staging if the
// toolchain doesn't declare the builtins).
#if defined(__AMDGCN__) && \
    __has_builtin(__builtin_amdgcn_global_load_async_to_lds_b128) && \
    __has_builtin(__builtin_amdgcn_s_wait_asynccnt)
#define MHA_ASYNC 1
#else
#define MHA_ASYNC 0
#endif

#if MHA_ASYNC
// Builtin signature (from hipcc diagnostic): arg0 is AS1 ("__device__") int4*,
// arg1 is the LDS-side int4*, then imm offset + imm cpol.
typedef __attribute__((address_space(1))) v4i* mha_gptr4;
typedef __attribute__((address_space(3))) v4i* mha_lptr4;

__device__ __forceinline__ void mha_async_b128(const void* g, void* l)
{
    __builtin_amdgcn_global_load_async_to_lds_b128(
        (mha_gptr4)g, (mha_lptr4)l, /*imm offset=*/0, /*cpol=*/0);
}
#endif

// ---------------------------------------------------------------------------
// Generic tiled GEMM:  C[M,N] = A[M,K] @ B[K,N] (+ bias), fp32 WMMA 16x16x4.
// 128 threads (4 waves, wave32). Block tile 64x64, K-chunk 32, double-buffered
// LDS fed by GLOBAL_LOAD_ASYNC_TO_LDS_B128 (ASYNCcnt pipeline).
// out_mode: 0 = plain row-major [M,N]
//           1 = head-split: m->(b,s), n->(h,dk), write [B,H,S,DK]
//           2 = merged attn-out: batch z=(b*H+h), write [B,S,D] at col h*DK+n
// ---------------------------------------------------------------------------
#define GM_BM 64
#define GM_BN 64
#define GM_BK 32

__global__ __launch_bounds__(128)
void mha_gemm_wmma_f32(const float* __restrict__ A,
                       const float* __restrict__ Bm,
                       const float* __restrict__ bias,
                       float* __restrict__ out,
                       int M, int N, int K,
                       long long strideA, long long strideB,
                       int out_mode)
{
    // Row strides 36/68 floats: rows stay 16B-aligned for B128 async stores.
    __shared__ float As[2][GM_BM][GM_BK + 4];
    __shared__ float Bs[2][GM_BK][GM_BN + 4];

    const int tid  = threadIdx.x;
    const int wave = tid >> 5;
    const int lane = tid & 31;
    const int half = lane >> 4;   // K/M half selector per ISA fragment layout
    const int lr   = lane & 15;

    const int z      = blockIdx.z;
    const int tile_m = blockIdx.y * GM_BM;
    const int tile_n = blockIdx.x * GM_BN;

    const float* Ab = A  + (long long)z * strideA;
    const float* Bb = Bm + (long long)z * strideB;

    // Stage one K-chunk into LDS buffer `buf`.
    auto stage = [&](int buf, int kb) {
#if MHA_ASYNC
#pragma unroll
        for (int i = tid; i < GM_BM * GM_BK / 4; i += 128) {
            int r = i >> 3, c4 = (i & 7) * 4;             // 64 x (8 float4)
            mha_async_b128(&Ab[(long long)(tile_m + r) * K + kb + c4],
                           &As[buf][r][c4]);
        }
#pragma unroll
        for (int i = tid; i < GM_BK * GM_BN / 4; i += 128) {
            int r = i >> 4, c4 = (i & 15) * 4;            // 32 x (16 float4)
            mha_async_b128(&Bb[(long long)(kb + r) * N + tile_n + c4],
                           &Bs[buf][r][c4]);
        }
#else
#pragma unroll
        for (int i = tid; i < GM_BM * GM_BK / 4; i += 128) {
            int r = i >> 3, c4 = (i & 7) * 4;
            float4 v = *reinterpret_cast<const float4*>(
                &Ab[(long long)(tile_m + r) * K + kb + c4]);
            As[buf][r][c4 + 0] = v.x; As[buf][r][c4 + 1] = v.y;
            As[buf][r][c4 + 2] = v.z; As[buf][r][c4 + 3] = v.w;
        }
#pragma unroll
        for (int i = tid; i < GM_BK * GM_BN / 4; i += 128) {
            int r = i >> 4, c4 = (i & 15) * 4;
            float4 v = *reinterpret_cast<const float4*>(
                &Bb[(long long)(kb + r) * N + tile_n + c4]);
            Bs[buf][r][c4 + 0] = v.x; Bs[buf][r][c4 + 1] = v.y;
            Bs[buf][r][c4 + 2] = v.z; Bs[buf][r][c4 + 3] = v.w;
        }
#endif
    };

    const v8f vzero = {0.f,0.f,0.f,0.f,0.f,0.f,0.f,0.f};
    v8f acc[4];
#pragma unroll
    for (int i = 0; i < 4; ++i) acc[i] = vzero;

    const int nk = K / GM_BK;
    stage(0, 0);

    for (int it = 0; it < nk; ++it) {
        const int buf = it & 1;
#if MHA_ASYNC
        __builtin_amdgcn_s_wait_asynccnt(0);
#endif
        // Single barrier per chunk: proves buffer `buf` is staged AND that all
        // waves finished reading buffer buf^1 (so it is safe to overwrite).
        __syncthreads();
        if (it + 1 < nk) stage(buf ^ 1, (it + 1) * GM_BK);

#pragma unroll
        for (int k = 0; k < GM_BK; k += 4) {
            // A fragment: lane holds A[row][k + 2*half + {0,1}] (contiguous)
            v2f a = *reinterpret_cast<const v2f*>(
                &As[buf][wave * 16 + lr][k + 2 * half]);
#pragma unroll
            for (int ns = 0; ns < 4; ++ns) {
                v2f b;
                b.x = Bs[buf][k + 2 * half + 0][ns * 16 + lr];
                b.y = Bs[buf][k + 2 * half + 1][ns * 16 + lr];
                acc[ns] = __builtin_amdgcn_wmma_f32_16x16x4_f32(
                    false, a, false, b, (short)0, acc[ns], false, false);
            }
        }
#if !MHA_ASYNC
        __syncthreads();   // sync path: order LDS reads before next overwrite
#endif
    }

    // Epilogue: C VGPR r -> row (8*half + r), col lr within 16x16 subtile
#pragma unroll
    for (int ns = 0; ns < 4; ++ns) {
#pragma unroll
        for (int r = 0; r < 8; ++r) {
            int m = tile_m + wave * 16 + half * 8 + r;
            int n = tile_n + ns * 16 + lr;
            float val = acc[ns][r];
            if (bias) val += bias[n];
            if (out_mode == 0) {
                out[(long long)m * N + n] = val;
            } else if (out_mode == 1) {
                int bb = m / Sc, s = m % Sc, h = n / DKc, dk = n % DKc;
                out[(((long long)(bb * Hc + h)) * Sc + s) * DKc + dk] = val;
            } else {
                int bb = z / Hc, h = z % Hc;
                out[((long long)bb * Sc + m) * Dc + h * DKc + n] = val;
            }
        }
    }
}

// ---------------------------------------------------------------------------
// Attention scores: per (b,h) 64x64 tile of (Q @ K^T) * 1/sqrt(DK), masked
// (mask==true -> -inf), written to the attn_weights slab of d_out.
// The B-fragment of K^T is exactly an A-style fragment of K (float2 LDS read).
// ---------------------------------------------------------------------------
__global__ __launch_bounds__(128)
void mha_scores_wmma_f32(const float* __restrict__ Qh,
                         const float* __restrict__ Kh,
                         const unsigned char* __restrict__ mask,
                         float* __restrict__ scores)
{
    __shared__ float Qs[64][DKc + 4];
    __shared__ float Ks[64][DKc + 4];

    const int tid  = threadIdx.x;
    const int wave = tid >> 5;
    const int lane = tid & 31;
    const int half = lane >> 4;
    const int lr   = lane & 15;

    const int z  = blockIdx.z;          // b*H + h
    const int ti = blockIdx.y * 64;     // query-row tile
    const int tj = blockIdx.x * 64;     // key-row tile

    const float* Qb = Qh + (long long)z * Sc * DKc;
    const float* Kb = Kh + (long long)z * Sc * DKc;

#if MHA_ASYNC
#pragma unroll
    for (int i = tid; i < 64 * DKc / 4; i += 128) {
        int r = i >> 4, c4 = (i & 15) * 4;                // 64 x (16 float4)
        mha_async_b128(&Qb[(long long)(ti + r) * DKc + c4], &Qs[r][c4]);
        mha_async_b128(&Kb[(long long)(tj + r) * DKc + c4], &Ks[r][c4]);
    }
    __builtin_amdgcn_s_wait_asynccnt(0);
#else
#pragma unroll
    for (int i = tid; i < 64 * DKc / 4; i += 128) {
        int r = i >> 4, c4 = (i & 15) * 4;
        float4 q = *reinterpret_cast<const float4*>(&Qb[(long long)(ti + r) * DKc + c4]);
        Qs[r][c4 + 0] = q.x; Qs[r][c4 + 1] = q.y; Qs[r][c4 + 2] = q.z; Qs[r][c4 + 3] = q.w;
        float4 k = *reinterpret_cast<const float4*>(&Kb[(long long)(tj + r) * DKc + c4]);
        Ks[r][c4 + 0] = k.x; Ks[r][c4 + 1] = k.y; Ks[r][c4 + 2] = k.z; Ks[r][c4 + 3] = k.w;
    }
#endif
    __syncthreads();

    const v8f vzero = {0.f,0.f,0.f,0.f,0.f,0.f,0.f,0.f};
    v8f acc[4];
#pragma unroll
    for (int i = 0; i < 4; ++i) acc[i] = vzero;

#pragma unroll
    for (int k = 0; k < DKc; k += 4) {
        v2f a = *reinterpret_cast<const v2f*>(&Qs[wave * 16 + lr][k + 2 * half]);
#pragma unroll
        for (int ns = 0; ns < 4; ++ns) {
            // B = K^T: B[k][n] = K[n][k] -> contiguous float2 from K's row
            v2f b = *reinterpret_cast<const v2f*>(&Ks[ns * 16 + lr][k + 2 * half]);
            acc[ns] = __builtin_amdgcn_wmma_f32_16x16x4_f32(
                false, a, false, b, (short)0, acc[ns], false, false);
        }
    }

    const float scale = 0.125f;  // 1/sqrt(64)
#pragma unroll
    for (int ns = 0; ns < 4; ++ns) {
#pragma unroll
        for (int r = 0; r < 8; ++r) {
            int i = ti + wave * 16 + half * 8 + r;
            int j = tj + ns * 16 + lr;
            long long idx = ((long long)z * Sc + i) * Sc + j;
            float val = acc[ns][r] * scale;
            if (mask[idx]) val = -INFINITY;
            scores[idx] = val;
        }
    }
}

// ---------------------------------------------------------------------------
// In-place row softmax over S=1024 columns. One 256-thread block per row.
// ---------------------------------------------------------------------------
__global__ __launch_bounds__(256)
void mha_softmax_rows(float* __restrict__ attn)
{
    const long long row = blockIdx.x;
    float* p = attn + row * (long long)Sc;
    const int tid = threadIdx.x;

    float x[4];
    float m = -INFINITY;
#pragma unroll
    for (int i = 0; i < 4; ++i) {
        x[i] = p[tid + i * 256];
        m = fmaxf(m, x[i]);
    }
#pragma unroll
    for (int off = 16; off > 0; off >>= 1)
        m = fmaxf(m, __shfl_down(m, off, 32));

    __shared__ float redm[8];
    __shared__ float reds[8];
    if ((tid & 31) == 0) redm[tid >> 5] = m;
    __syncthreads();
    float mall = redm[0];
#pragma unroll
    for (int i = 1; i < 8; ++i) mall = fmaxf(mall, redm[i]);

    float s = 0.f;
#pragma unroll
    for (int i = 0; i < 4; ++i) {
        x[i] = expf(x[i] - mall);
        s += x[i];
    }
#pragma unroll
    for (int off = 16; off > 0; off >>= 1)
        s += __shfl_down(s, off, 32);
    if ((tid & 31) == 0) reds[tid >> 5] = s;
    __syncthreads();
    float stot = 0.f;
#pragma unroll
    for (int i = 0; i < 8; ++i) stot += reds[i];

    const float inv = 1.0f / stot;
#pragma unroll
    for (int i = 0; i < 4; ++i)
        p[tid + i * 256] = x[i] * inv;
}

// ---------------------------------------------------------------------------
extern "C" void kernel_launch(void* const* d_in, const int* in_sizes, int n_in,
                              void* d_out, int out_size, void* d_ws, size_t ws_size,
                              hipStream_t stream)
{
    (void)in_sizes; (void)n_in; (void)out_size; (void)ws_size;

    const float* query = (const float*)d_in[0];
    const float* key   = (const float*)d_in[1];
    const float* value = (const float*)d_in[2];
    const unsigned char* mask = (const unsigned char*)d_in[3];  // jnp bool -> 1B
    const float* Wq = (const float*)d_in[4];
    const float* bq = (const float*)d_in[5];
    const float* Wk = (const float*)d_in[6];
    const float* bk = (const float*)d_in[7];
    const float* Wv = (const float*)d_in[8];
    const float* bv = (const float*)d_in[9];
    const float* Wo = (const float*)d_in[10];
    const float* bo = (const float*)d_in[11];

    const long long nBSD = (long long)Bc * Sc * Dc;          // 4M
    float* out   = (float*)d_out;                            // [B,S,D]
    float* attnw = out + nBSD;                               // [B,H,S,S]

    float* ws = (float*)d_ws;
    float* Qh = ws;                 // [B,H,S,DK]
    float* Kh = ws + nBSD;          // [B,H,S,DK]
    float* Vh = ws + 2 * nBSD;      // [B,H,S,DK]
    float* AO = ws + 3 * nBSD;      // merged attn out [B,S,D]

    dim3 blk(128);

    // QKV projections: [4096,1024] @ [1024,1024] + bias -> head-split
    dim3 g_proj(Dc / GM_BN, (Bc * Sc) / GM_BM, 1);
    mha_gemm_wmma_f32<<<g_proj, blk, 0, stream>>>(query, Wq, bq, Qh,
        Bc * Sc, Dc, Dc, 0, 0, /*out_mode=*/1);
    mha_gemm_wmma_f32<<<g_proj, blk, 0, stream>>>(key, Wk, bk, Kh,
        Bc * Sc, Dc, Dc, 0, 0, /*out_mode=*/1);
    mha_gemm_wmma_f32<<<g_proj, blk, 0, stream>>>(value, Wv, bv, Vh,
        Bc * Sc, Dc, Dc, 0, 0, /*out_mode=*/1);

    // Scores (scaled, masked) -> attn_weights slab of d_out
    dim3 g_sc(Sc / 64, Sc / 64, Bc * Hc);
    mha_scores_wmma_f32<<<g_sc, blk, 0, stream>>>(Qh, Kh, mask, attnw);

    // Row softmax in place -> final attn_weights
    mha_softmax_rows<<<dim3(Bc * Hc * Sc), dim3(256), 0, stream>>>(attnw);

    // attn_out = P @ V per (b,h): batched [1024,1024] @ [1024,64] -> merged [B,S,D]
    dim3 g_av(1, Sc / GM_BM, Bc * Hc);
    mha_gemm_wmma_f32<<<g_av, blk, 0, stream>>>(attnw, Vh, nullptr, AO,
        Sc, DKc, Sc, (long long)Sc * Sc, (long long)Sc * DKc, /*out_mode=*/2);

    // Output projection: [4096,1024] @ [1024,1024] + bias -> plain [B,S,D]
    mha_gemm_wmma_f32<<<g_proj, blk, 0, stream>>>(AO, Wo, bo, out,
        Bc * Sc, Dc, Dc, 0, 0, /*out_mode=*/0);
}